// GraphMessageEmbeddings_60971355734504
// MI455X (gfx1250) — compile-verified
//
#include <hip/hip_runtime.h>
#include <hip/hip_bf16.h>

#define N_NODES  50000
#define N_EDGES  800000
#define N_GRAPHS 512
#define F_IN     128
#define DOUT     256
#define BN_EPS   1e-5f
#define SLOPE    0.01f

typedef float v2f __attribute__((ext_vector_type(2)));
typedef float v8f __attribute__((ext_vector_type(8)));

__device__ __forceinline__ float leaky(float v) { return v >= 0.f ? v : SLOPE * v; }

// ---------------------------------------------------------------- fill
__global__ void fill_u32_kernel(unsigned int* p, unsigned int val, size_t n) {
    size_t i = (size_t)blockIdx.x * blockDim.x + threadIdx.x;
    size_t stride = (size_t)gridDim.x * blockDim.x;
    for (; i < n; i += stride) p[i] = val;
}

// ---------------------------------------------------------------- fuse weights
// wc[d][j] = sum_k lin_w[d][k] * lin2_w[k][j]   (j=0..3)
// bvec[d]  = sum_k lin_w[d][k] * lin2_b[k]
__global__ void fuse_w_kernel(const float* __restrict__ lin_w,
                              const float* __restrict__ lin2_w,
                              const float* __restrict__ lin2_b,
                              float* __restrict__ wc, float* __restrict__ bvec, int K) {
    int d = blockIdx.x * blockDim.x + threadIdx.x;
    if (d >= DOUT) return;
    float a0 = 0.f, a1 = 0.f, a2 = 0.f, a3 = 0.f, bv = 0.f;
    const float* wr = lin_w + (size_t)d * K;
    for (int k = 0; k < K; ++k) {
        float w = wr[k];
        bv += w * lin2_b[k];
        a0 += w * lin2_w[k * 4 + 0];
        a1 += w * lin2_w[k * 4 + 1];
        a2 += w * lin2_w[k * 4 + 2];
        a3 += w * lin2_w[k * 4 + 3];
    }
    wc[d * 4 + 0] = a0; wc[d * 4 + 1] = a1;
    wc[d * 4 + 2] = a2; wc[d * 4 + 3] = a3;
    bvec[d] = bv;
}

// ---------------------------------------------------------------- WMMA GEMM
// Out[m][n] = sum_k A[m][k] * W[n][k] + bias[n]
// A row m comes from Asrc[ids[m]] when ids != nullptr (embedding gather),
// else Asrc[m]. One wave computes one 16x16 tile via v_wmma_f32_16x16x4_f32.
__global__ void gemm_wmma_f32_kernel(const float* __restrict__ Asrc,
                                     const int* __restrict__ ids,
                                     const float* __restrict__ W,
                                     const float* __restrict__ bias,
                                     float* __restrict__ Out, int K) {
    int wave = blockIdx.x * (blockDim.x >> 5) + (threadIdx.x >> 5);
    int lane = threadIdx.x & 31;
    const int tilesN = DOUT / 16;
    int tm = wave / tilesN;
    int tn = wave % tilesN;
    int halfk = lane >> 4;   // 0: K pair {0,1}; 1: K pair {2,3}
    int lr = lane & 15;
    int m = tm * 16 + lr;    // A row this lane supplies
    int n = tn * 16 + lr;    // B column this lane supplies

    const float* arow = ids ? (Asrc + (size_t)ids[m] * K) : (Asrc + (size_t)m * K);
    const float* wrow = W + (size_t)n * K;

    v8f c;
    #pragma unroll
    for (int i = 0; i < 8; ++i) c[i] = 0.f;

    for (int k = 0; k < K; k += 4) {
        v2f a, b;
        // A 16x4 layout: lanes 0-15 hold K={k,k+1}, lanes 16-31 hold K={k+2,k+3}
        a[0] = arow[k + 2 * halfk + 0];
        a[1] = arow[k + 2 * halfk + 1];
        // B 4x16 layout mirrors A's K mapping, N across lanes
        b[0] = wrow[k + 2 * halfk + 0];
        b[1] = wrow[k + 2 * halfk + 1];
        c = __builtin_amdgcn_wmma_f32_16x16x4_f32(false, a, false, b,
                                                  (short)0, c, false, false);
    }

    // C/D layout: VGPR r -> M = r (lanes 0-15) / r+8 (lanes 16-31), N = lane&15
    int col = tn * 16 + lr;
    float bb = bias[col];
    #pragma unroll
    for (int r = 0; r < 8; ++r) {
        int row = tm * 16 + r + halfk * 8;
        Out[(size_t)row * DOUT + col] = c[r] + bb;
    }
}

// ---------------------------------------------------------------- edge scatter
// acc[dst] += leaky( h[src] + bvec + Wc * edge_attr[e] ); one block per edge.
__global__ void edge_scatter_kernel(const int* __restrict__ ei,
                                    const float* __restrict__ eattr,
                                    const float* __restrict__ h,
                                    const float* __restrict__ wc,
                                    const float* __restrict__ bvec,
                                    float* __restrict__ acc) {
    int e = blockIdx.x;
    int d = threadIdx.x;               // 0..255
    int s = ei[e];
    int t = ei[N_EDGES + e];
    float a0 = eattr[e * 4 + 0], a1 = eattr[e * 4 + 1];
    float a2 = eattr[e * 4 + 2], a3 = eattr[e * 4 + 3];
    const float* w = wc + d * 4;
    float v = h[(size_t)s * DOUT + d] + bvec[d]
            + w[0] * a0 + w[1] * a1 + w[2] * a2 + w[3] * a3;
    v = leaky(v);
    atomicAdd(&acc[(size_t)t * DOUT + d], v);
}

// ---------------------------------------------------------------- batch norm
__global__ void bn_partial_kernel(const float* __restrict__ x,
                                  float* __restrict__ sums,
                                  float* __restrict__ sumsq) {
    int c = threadIdx.x;               // channel
    int per = N_NODES / gridDim.x;     // grid chosen to divide exactly
    int n0 = blockIdx.x * per;
    int n1 = n0 + per;
    float s = 0.f, ss = 0.f;
    for (int n = n0; n < n1; ++n) {
        float v = x[(size_t)n * DOUT + c];
        s += v; ss += v * v;
    }
    atomicAdd(&sums[c], s);
    atomicAdd(&sumsq[c], ss);
}

__global__ void bn_apply_kernel(float* __restrict__ x,
                                const float* __restrict__ sums,
                                const float* __restrict__ sumsq,
                                const float* __restrict__ w,
                                const float* __restrict__ b) {
    size_t idx = (size_t)blockIdx.x * blockDim.x + threadIdx.x;
    int c = (int)(idx & (DOUT - 1));
    float mean = sums[c] * (1.0f / N_NODES);
    float var = sumsq[c] * (1.0f / N_NODES) - mean * mean;
    float rstd = rsqrtf(var + BN_EPS);
    float v = (x[idx] - mean) * rstd * w[c] + b[c];
    x[idx] = leaky(v);
}

// ---------------------------------------------------------------- gate
__global__ void gate_kernel(const float* __restrict__ x2,
                            const float* __restrict__ gw,
                            const float* __restrict__ gb,
                            float* __restrict__ gate) {
    int wave = (blockIdx.x * blockDim.x + threadIdx.x) >> 5;
    int lane = threadIdx.x & 31;
    float s = 0.f;
    const float* row = x2 + (size_t)wave * DOUT;
    for (int k = lane; k < DOUT; k += 32) s += row[k] * gw[k];
    #pragma unroll
    for (int off = 16; off >= 1; off >>= 1) s += __shfl_xor(s, off, 32);
    if (lane == 0) gate[wave] = s + gb[0];
}

// ---------------------------------------------------------------- segment softmax
__device__ __forceinline__ unsigned int map_f32(float f) {
    unsigned int u = __float_as_uint(f);
    return (u & 0x80000000u) ? ~u : (u | 0x80000000u);
}
__device__ __forceinline__ float unmap_f32(unsigned int m) {
    return __uint_as_float((m & 0x80000000u) ? (m & 0x7FFFFFFFu) : ~m);
}

__global__ void seg_max_kernel(const float* __restrict__ gate,
                               const int* __restrict__ batch,
                               unsigned int* __restrict__ gmax) {
    int n = blockIdx.x * blockDim.x + threadIdx.x;
    if (n >= N_NODES) return;
    atomicMax(&gmax[batch[n]], map_f32(gate[n]));
}

__global__ void seg_expsum_kernel(const float* __restrict__ gate,
                                  const int* __restrict__ batch,
                                  const unsigned int* __restrict__ gmax,
                                  float* __restrict__ ebuf,
                                  float* __restrict__ denom) {
    int n = blockIdx.x * blockDim.x + threadIdx.x;
    if (n >= N_NODES) return;
    int g = batch[n];
    float e = __expf(gate[n] - unmap_f32(gmax[g]));
    ebuf[n] = e;
    atomicAdd(&denom[g], e);
}

__global__ void seg_pool_kernel(const float* __restrict__ x2,
                                const int* __restrict__ batch,
                                const float* __restrict__ ebuf,
                                const float* __restrict__ denom,
                                float* __restrict__ out) {
    int n = blockIdx.x;
    int d = threadIdx.x;
    int g = batch[n];
    float alpha = ebuf[n] / denom[g];
    atomicAdd(&out[(size_t)g * DOUT + d], alpha * x2[(size_t)n * DOUT + d]);
}

// ---------------------------------------------------------------- launch
extern "C" void kernel_launch(void* const* d_in, const int* in_sizes, int n_in,
                              void* d_out, int out_size, void* d_ws, size_t ws_size,
                              hipStream_t stream) {
    const int*   node_ids  = (const int*)d_in[0];
    const int*   edge_idx  = (const int*)d_in[1];
    const float* edge_attr = (const float*)d_in[2];
    const int*   batch     = (const int*)d_in[3];
    const float* emb       = (const float*)d_in[4];
    const float* c1_l2w    = (const float*)d_in[5];
    const float* c1_l2b    = (const float*)d_in[6];
    const float* c1_lw     = (const float*)d_in[7];
    const float* c1_lb     = (const float*)d_in[8];
    const float* bn1_w     = (const float*)d_in[9];
    const float* bn1_b     = (const float*)d_in[10];
    const float* c2_l2w    = (const float*)d_in[11];
    const float* c2_l2b    = (const float*)d_in[12];
    const float* c2_lw     = (const float*)d_in[13];
    const float* c2_lb     = (const float*)d_in[14];
    const float* bn2_w     = (const float*)d_in[15];
    const float* bn2_b     = (const float*)d_in[16];
    const float* gate_w    = (const float*)d_in[17];
    const float* gate_b    = (const float*)d_in[18];
    float* out = (float*)d_out;

    // workspace carve-up
    char* base = (char*)d_ws;
    const size_t NODE_MAT = (size_t)N_NODES * DOUT * sizeof(float); // 51.2 MB
    float* hbuf = (float*)(base);                         // h1 then h2
    float* xbuf = (float*)(base + NODE_MAT);              // acc1->x1, acc2->x2
    char*  sm   = base + 2 * NODE_MAT;
    float* wc1   = (float*)(sm);            sm += DOUT * 4 * sizeof(float);
    float* wc2   = (float*)(sm);            sm += DOUT * 4 * sizeof(float);
    float* bvec1 = (float*)(sm);            sm += DOUT * sizeof(float);
    float* bvec2 = (float*)(sm);            sm += DOUT * sizeof(float);
    float* sums  = (float*)(sm);            sm += DOUT * sizeof(float);
    float* sumsq = (float*)(sm);            sm += DOUT * sizeof(float);
    float* gbuf  = (float*)(sm);            sm += N_NODES * sizeof(float);
    float* ebuf  = (float*)(sm);            sm += N_NODES * sizeof(float);
    unsigned int* gmax = (unsigned int*)(sm); sm += N_GRAPHS * sizeof(unsigned int);
    float* denom = (float*)(sm);            sm += N_GRAPHS * sizeof(float);

    const size_t NODE_ELEMS = (size_t)N_NODES * DOUT;
    const int TILES  = (N_NODES / 16) * (DOUT / 16);      // 50000 waves
    const int GBLK   = TILES / 4;                          // 4 waves / block
    const unsigned NEG_INF_MAPPED = 0x007FFFFFu;           // map_f32(-inf)

    // ---- init
    fill_u32_kernel<<<2048, 256, 0, stream>>>((unsigned int*)xbuf, 0u, NODE_ELEMS);
    fill_u32_kernel<<<2, 256, 0, stream>>>((unsigned int*)sums, 0u, DOUT);
    fill_u32_kernel<<<2, 256, 0, stream>>>((unsigned int*)sumsq, 0u, DOUT);
    fill_u32_kernel<<<512, 256, 0, stream>>>((unsigned int*)out, 0u, (size_t)N_GRAPHS * DOUT);
    fill_u32_kernel<<<2, 256, 0, stream>>>((unsigned int*)denom, 0u, N_GRAPHS);
    fill_u32_kernel<<<2, 256, 0, stream>>>(gmax, NEG_INF_MAPPED, N_GRAPHS);

    // ---- fused edge-path weights: Wc = lin_w @ lin2_w, bvec = lin_w @ lin2_b
    fuse_w_kernel<<<1, 256, 0, stream>>>(c1_lw, c1_l2w, c1_l2b, wc1, bvec1, F_IN);
    fuse_w_kernel<<<1, 256, 0, stream>>>(c2_lw, c2_l2w, c2_l2b, wc2, bvec2, DOUT);

    // ---- layer 1: h1 = gather(emb, node_ids) @ W1^T + b1   (WMMA f32)
    gemm_wmma_f32_kernel<<<GBLK, 128, 0, stream>>>(emb, node_ids, c1_lw, c1_lb, hbuf, F_IN);
    // edge scatter-add into xbuf
    edge_scatter_kernel<<<N_EDGES, DOUT, 0, stream>>>(edge_idx, edge_attr, hbuf, wc1, bvec1, xbuf);
    // BN1 + leaky (in place)
    bn_partial_kernel<<<250, DOUT, 0, stream>>>(xbuf, sums, sumsq);
    bn_apply_kernel<<<(int)(NODE_ELEMS / 256), 256, 0, stream>>>(xbuf, sums, sumsq, bn1_w, bn1_b);

    // ---- layer 2: h2 = x1 @ W2^T + b2   (WMMA f32)
    gemm_wmma_f32_kernel<<<GBLK, 128, 0, stream>>>(xbuf, nullptr, c2_lw, c2_lb, hbuf, DOUT);
    // reset accumulators / stats, then edge scatter-add into xbuf
    fill_u32_kernel<<<2048, 256, 0, stream>>>((unsigned int*)xbuf, 0u, NODE_ELEMS);
    fill_u32_kernel<<<2, 256, 0, stream>>>((unsigned int*)sums, 0u, DOUT);
    fill_u32_kernel<<<2, 256, 0, stream>>>((unsigned int*)sumsq, 0u, DOUT);
    edge_scatter_kernel<<<N_EDGES, DOUT, 0, stream>>>(edge_idx, edge_attr, hbuf, wc2, bvec2, xbuf);
    // BN2 + leaky (in place) -> x2
    bn_partial_kernel<<<250, DOUT, 0, stream>>>(xbuf, sums, sumsq);
    bn_apply_kernel<<<(int)(NODE_ELEMS / 256), 256, 0, stream>>>(xbuf, sums, sumsq, bn2_w, bn2_b);

    // ---- attentional aggregation
    gate_kernel<<<N_NODES / 8, 256, 0, stream>>>(xbuf, gate_w, gate_b, gbuf);
    seg_max_kernel<<<(N_NODES + 255) / 256, 256, 0, stream>>>(gbuf, batch, gmax);
    seg_expsum_kernel<<<(N_NODES + 255) / 256, 256, 0, stream>>>(gbuf, batch, gmax, ebuf, denom);
    seg_pool_kernel<<<N_NODES, DOUT, 0, stream>>>(xbuf, batch, ebuf, denom, out);
}